// ResBlock1_65403761984293
// MI455X (gfx1250) — compile-verified
//
#include <hip/hip_runtime.h>

typedef __attribute__((ext_vector_type(2))) float v2f;
typedef __attribute__((ext_vector_type(8))) float v8f;

#define T_LEN   8192
#define B_SZ    8
#define C_IN    192
#define C_INNER 384
#define C_OUT   192
#define G_IN    256
#define R_LORA  4
#define NCOL    64            // columns per tile (incl. halo)
#define HALO    18            // 2*(k-1)*max-chain: 2*1 + 2*3 + 2*5
#define TT      (NCOL - HALO) // 46 useful output columns per tile
#define LDP     65            // padded LDS row stride (floats) - avoids bank conflicts
#define NTHR    512
#define LRELU_S 0.1f
#define EPS_LN  1e-5f

// ---------------------------------------------------------------------------
// Phase 1: fold LoRA into effective per-batch weights
//   a_in[b]  = reshape(wai @ g[b] + bai, (cin, R))
//   a_out[b] = reshape(wao @ g[b] + bao, (R, cout))
//   W_eff[b][o][c] = wm[o][c] + sum_r a_out[b][r][o] * a_in[b][c][r]
// ---------------------------------------------------------------------------
struct PrepArgs {
  const float* g;             // (B, GIN)
  const float* wm[4];
  const float* wai[4];
  const float* bai[4];
  const float* wao[4];
  const float* bao[4];
  float*       weff[4];       // scratch out: [b][cout][cin] per layer
  int cin[4];
  int cout[4];
};

__global__ __launch_bounds__(256) void prep_weff(PrepArgs p) {
  __shared__ float sg[G_IN];
  __shared__ float sain[C_INNER * R_LORA];   // max cin*R  = 1536
  __shared__ float saout[C_INNER * R_LORA];  // max cout*R = 1536
  const int lay = blockIdx.x >> 3;
  const int b   = blockIdx.x & 7;
  const int tid = threadIdx.x;
  const int cin = p.cin[lay], cout = p.cout[lay];

  for (int i = tid; i < G_IN; i += 256) sg[i] = p.g[b * G_IN + i];
  __syncthreads();

  const int nin = cin * R_LORA, nout = cout * R_LORA;
  for (int o = tid; o < nin; o += 256) {
    float acc = p.bai[lay][o];
    const float* w = p.wai[lay] + (size_t)o * G_IN;
    for (int k = 0; k < G_IN; ++k) acc += w[k] * sg[k];
    sain[o] = acc;
  }
  for (int o = tid; o < nout; o += 256) {
    float acc = p.bao[lay][o];
    const float* w = p.wao[lay] + (size_t)o * G_IN;
    for (int k = 0; k < G_IN; ++k) acc += w[k] * sg[k];
    saout[o] = acc;
  }
  __syncthreads();

  const int n = cout * cin;
  float* dst = p.weff[lay] + (size_t)b * n;
  for (int idx = tid; idx < n; idx += 256) {
    const int m = idx / cin, k = idx - m * cin;
    float acc = p.wm[lay][idx];
#pragma unroll
    for (int r = 0; r < R_LORA; ++r)
      acc += saout[r * cout + m] * sain[k * R_LORA + r];
    dst[idx] = acc;
  }
}

// ---------------------------------------------------------------------------
// Per-tile GEMM: C(Mtot x 64) = W(Mtot x Kc) * B(Kc x 64 in LDS) [+ Cinit] + bias
// wave w: N-block = w&3 (16 cols), M-group = w>>2 owning MB consecutive 16-row
// blocks. K stepped 4 at a time with V_WMMA_F32_16X16X4_F32.
// A frag layout (16x4 f32): VGPR i, lanes 0-15 -> K=i, lanes 16-31 -> K=2+i.
// C/D layout: VGPR j, lanes 0-15 -> M=j, lanes 16-31 -> M=j+8; N=lane&15.
// ---------------------------------------------------------------------------
template<int MB, bool ADDX, bool TOGLOB>
__device__ __forceinline__ void gemm_tile(
    const float* __restrict__ W, int Kc, const float* __restrict__ bias,
    const float* Bsrc, const float* Cinit, float* Cdst,
    float* gout, int t0)
{
  const int tid  = threadIdx.x;
  const int lane = tid & 31;
  const int wave = tid >> 5;
  const int nb   = wave & 3;
  const int mg   = wave >> 2;
  const int colB = nb * 16 + (lane & 15);
  const int hk   = (lane >> 4) << 1;     // 0 or 2 (K-half per lane group)
  const int rowl = lane & 15;

  v8f acc[MB];
#pragma unroll
  for (int i = 0; i < MB; ++i) {
    const int mbase = (mg * MB + i) * 16 + ((lane >> 4) << 3);
#pragma unroll
    for (int j = 0; j < 8; ++j)
      acc[i][j] = ADDX ? Cinit[(mbase + j) * LDP + colB] : 0.0f;
  }

  const int nk = Kc >> 2;
  for (int kb = 0; kb < nk; ++kb) {
    const int kcol = (kb << 2) + hk;
    v2f bf;
    bf.x = Bsrc[kcol * LDP + colB];
    bf.y = Bsrc[(kcol + 1) * LDP + colB];
#pragma unroll
    for (int i = 0; i < MB; ++i) {
      const int row = (mg * MB + i) * 16 + rowl;
      const float* ap = W + (size_t)row * Kc + kcol;
      v2f af; af.x = ap[0]; af.y = ap[1];
      acc[i] = __builtin_amdgcn_wmma_f32_16x16x4_f32(
          false, af, false, bf, (short)0, acc[i], false, false);
    }
  }

#pragma unroll
  for (int i = 0; i < MB; ++i) {
    const int mbase = (mg * MB + i) * 16 + ((lane >> 4) << 3);
#pragma unroll
    for (int j = 0; j < 8; ++j) {
      const int m = mbase + j;
      const float v = acc[i][j] + bias[m];
      if (TOGLOB) {
        const int t = t0 + colB - HALO;
        if (colB >= HALO && t < T_LEN)
          gout[(size_t)m * T_LEN + t] = v;
      } else {
        Cdst[m * LDP + colB] = v;
      }
    }
  }
}

// ---------------------------------------------------------------------------
// Phase 2: fully fused resblock over one (batch, time-tile)
// ---------------------------------------------------------------------------
struct FusedArgs {
  const float* x;        // (B, 192, T)
  const float* mask;     // (B, 1, T)
  const float* weff[4];  // [0]=init 384x192, [1]=p0, [2]=p1 384x384, [3]=out 192x384
  const float* bm[4];
  const float* dw[3];    // (384,1,3)
  const float* db[3];
  const float* gamma[3];
  const float* beta[3];
  float* out;            // (B, 192, T)
};

__global__ __launch_bounds__(NTHR, 1) void resblock_fused(FusedArgs a) {
  extern __shared__ float smem[];
  float* bufX = smem;                      // running residual x   (384 x 64)
  float* bufY = smem + C_INNER * LDP;      // layernorm output xn  (384 x 64)
  float* bufZ = smem + 2 * C_INNER * LDP;  // conv output x_ / X0  (384 x 64)

  const int tile = blockIdx.x;
  const int b    = blockIdx.y;
  const int t0   = tile * TT;
  const int tid  = threadIdx.x;

  const float* xin = a.x    + (size_t)b * C_IN  * T_LEN;
  const float* msk = a.mask + (size_t)b * T_LEN;
  float* gout      = a.out  + (size_t)b * C_OUT * T_LEN;

  // Load masked input columns [t0-18, t0+46) into bufZ (192 x 64)
  for (int idx = tid; idx < C_IN * NCOL; idx += NTHR) {
    const int ch = idx >> 6;
    const int e  = idx & 63;
    const int t  = t0 - HALO + e;
    float v = 0.0f;
    if (t >= 0 && t < T_LEN) v = xin[(size_t)ch * T_LEN + t] * msk[t];
    bufZ[ch * LDP + e] = v;
  }
  __syncthreads();

  // init conv: bufX = Weff0[b] @ X0 + bm0
  gemm_tile<6, false, false>(a.weff[0] + (size_t)b * C_INNER * C_IN, C_IN,
                             a.bm[0], bufZ, nullptr, bufX, nullptr, t0);
  __syncthreads();

  const int ez = (tile == 0) ? HALO : 0;   // conv taps with global t<0 read 0

#pragma unroll
  for (int s = 0; s < 3; ++s) {
    const int d = (s == 0) ? 1 : (s == 1 ? 3 : 5);

    // x *= mask (mask treated as 0 outside [0,T))
    for (int idx = tid; idx < C_INNER * NCOL; idx += NTHR) {
      const int ch = idx >> 6, e = idx & 63;
      const int t = t0 - HALO + e;
      const float mv = (t >= 0 && t < T_LEN) ? msk[t] : 0.0f;
      bufX[ch * LDP + e] *= mv;
    }
    __syncthreads();

    // channel layernorm -> bufY ; 8-lane team per column, wave32 shuffles
    {
      const int col = tid >> 3;     // 64 columns, all 512 threads active
      const int p   = tid & 7;
      float s1 = 0.0f, s2 = 0.0f;
      for (int ch = p; ch < C_INNER; ch += 8) {
        const float v = bufX[ch * LDP + col];
        s1 += v; s2 += v * v;
      }
      s1 += __shfl_xor(s1, 1); s2 += __shfl_xor(s2, 1);
      s1 += __shfl_xor(s1, 2); s2 += __shfl_xor(s2, 2);
      s1 += __shfl_xor(s1, 4); s2 += __shfl_xor(s2, 4);
      const float mean = s1 * (1.0f / C_INNER);
      const float var  = s2 * (1.0f / C_INNER) - mean * mean;
      const float rs   = rsqrtf(var + EPS_LN);
      const float* gam = a.gamma[s];
      const float* bet = a.beta[s];
      for (int ch = p; ch < C_INNER; ch += 8) {
        const float v = bufX[ch * LDP + col];
        bufY[ch * LDP + col] = (v - mean) * rs * gam[ch] + bet[ch];
      }
    }
    __syncthreads();

    // dilated causal depthwise conv + bias + leaky relu -> bufZ
    {
      const float* dw = a.dw[s];
      const float* db = a.db[s];
      for (int idx = tid; idx < C_INNER * NCOL; idx += NTHR) {
        const int ch = idx >> 6, e = idx & 63;
        float acc = db[ch];
#pragma unroll
        for (int j = 0; j < 3; ++j) {
          const int et = e - j * d;
          const float v = (et >= ez) ? bufY[ch * LDP + et] : 0.0f;
          acc += dw[ch * 3 + j] * v;
        }
        acc = fmaxf(acc, 0.0f) + LRELU_S * fminf(acc, 0.0f);
        bufZ[ch * LDP + e] = acc;
      }
    }
    __syncthreads();

    if (s < 2) {
      // x = x + Weff_pconv[b] @ x_ + bm   (accumulator seeded from bufX)
      gemm_tile<6, true, false>(a.weff[1 + s] + (size_t)b * C_INNER * C_INNER,
                                C_INNER, a.bm[1 + s], bufZ, bufX, bufX,
                                nullptr, t0);
      __syncthreads();
    } else {
      // out = Weff_out[b] @ x_ + bm_out  -> global (valid columns only)
      gemm_tile<3, false, true>(a.weff[3] + (size_t)b * C_OUT * C_INNER,
                                C_INNER, a.bm[3], bufZ, nullptr, nullptr,
                                gout, t0);
    }
  }
}

// ---------------------------------------------------------------------------
extern "C" void kernel_launch(void* const* d_in, const int* in_sizes, int n_in,
                              void* d_out, int out_size, void* d_ws, size_t ws_size,
                              hipStream_t stream) {
  (void)in_sizes; (void)n_in; (void)out_size; (void)ws_size;

  // d_in flatten order: x, x_mask, g,
  //  init_conv{wm,bm,wai,bai,wao,bao}=3..8, out_conv=9..14,
  //  pconvs[0]=15..20, pconvs[1]=21..26,
  //  dconvs{w,b}x3=27..32, norms{gamma,beta}x3=33..38
  const float* x    = (const float*)d_in[0];
  const float* mask = (const float*)d_in[1];
  const float* g    = (const float*)d_in[2];

  static const int base[4]  = {3, 15, 21, 9};          // init, pconv0, pconv1, out
  static const int cins[4]  = {C_IN, C_INNER, C_INNER, C_INNER};
  static const int couts[4] = {C_INNER, C_INNER, C_INNER, C_OUT};

  PrepArgs pa;
  pa.g = g;
  float* ws = (float*)d_ws;
  size_t off = 0;
  for (int l = 0; l < 4; ++l) {
    pa.wm[l]  = (const float*)d_in[base[l] + 0];
    pa.wai[l] = (const float*)d_in[base[l] + 2];
    pa.bai[l] = (const float*)d_in[base[l] + 3];
    pa.wao[l] = (const float*)d_in[base[l] + 4];
    pa.bao[l] = (const float*)d_in[base[l] + 5];
    pa.cin[l]  = cins[l];
    pa.cout[l] = couts[l];
    pa.weff[l] = ws + off;
    off += (size_t)B_SZ * cins[l] * couts[l];          // total 3,538,944 floats (~14.2 MB)
  }
  prep_weff<<<dim3(32), dim3(256), 0, stream>>>(pa);

  FusedArgs fa;
  fa.x = x; fa.mask = mask; fa.out = (float*)d_out;
  for (int l = 0; l < 4; ++l) {
    fa.weff[l] = pa.weff[l];
    fa.bm[l]   = (const float*)d_in[base[l] + 1];
  }
  for (int i = 0; i < 3; ++i) {
    fa.dw[i]    = (const float*)d_in[27 + 2 * i];
    fa.db[i]    = (const float*)d_in[28 + 2 * i];
    fa.gamma[i] = (const float*)d_in[33 + 2 * i];
    fa.beta[i]  = (const float*)d_in[34 + 2 * i];
  }

  const int ntile = (T_LEN + TT - 1) / TT;             // 179
  const size_t smem = (size_t)3 * C_INNER * LDP * sizeof(float); // 299,520 B < 320 KB
  resblock_fused<<<dim3(ntile, B_SZ), dim3(NTHR), smem, stream>>>(fa);
}